// DenSaE_72344429133962
// MI455X (gfx1250) — compile-verified
//
#include <hip/hip_runtime.h>
#include <cmath>

// ---------------- problem constants ----------------
#define NB 16
#define CH 32          // filters per dictionary (A and B each)
#define HY 256
#define WY 256
#define HX 250
#define WX 250
#define KD 7
#define KK 49          // 7*7
#define NITER 15

#define INV_L 0.1f     // 1/L
#define THR   0.01f    // LAM/L
#define XSCL  2.0f     // 1 + 1/RHO

static const long XSZ = (long)NB * CH * HX * WX;  // 32,000,000
static const long YSZ = (long)NB * HY * WY;       //  1,048,576

typedef float v2f __attribute__((ext_vector_type(2)));
typedef float v8f __attribute__((ext_vector_type(8)));

// =====================================================================
// Kernel R: res = y - ( convT(2*x_tmp, A) + convT(u_tmp, B) )
// x_tmp/u_tmp recomputed from (cur, prev, mom) at LDS-stage time.
// One 16x16 output tile per workgroup; loop over 32 channels, staging
// a 22x22 slice of x and u per channel into LDS.
// =====================================================================
__global__ __launch_bounds__(256) void densae_resid(
    float* __restrict__ res, const float* __restrict__ y,
    const float* __restrict__ A, const float* __restrict__ B,
    const float* __restrict__ xc, const float* __restrict__ xp,
    const float* __restrict__ uc, const float* __restrict__ up,
    float mom)
{
  __shared__ float swa[CH * KK];
  __shared__ float swb[CH * KK];
  __shared__ float sx[22][24];
  __shared__ float su[22][24];

  const int tx = threadIdx.x, ty = threadIdx.y;
  const int tid = ty * 16 + tx;
  const int n  = blockIdx.z;
  const int r0 = blockIdx.y * 16;
  const int c0 = blockIdx.x * 16;

  for (int i = tid; i < CH * KK; i += 256) { swa[i] = A[i]; swb[i] = B[i]; }

  const long nbase = (long)n * CH * HX * WX;
  float acc = 0.f;

  for (int f = 0; f < CH; ++f) {
    __syncthreads();
    const long cb = nbase + (long)f * HX * WX;
    for (int i = tid; i < 22 * 22; i += 256) {
      int rr = i / 22, cc = i % 22;
      int gi = r0 - 6 + rr, gj = c0 - 6 + cc;
      float vx = 0.f, vu = 0.f;
      if (gi >= 0 && gi < HX && gj >= 0 && gj < WX) {
        long idx = cb + (long)gi * WX + gj;
        float c1 = xc[idx], p1 = xp[idx];
        vx = XSCL * (c1 + mom * (c1 - p1));        // 2 * x_tmp
        float c2 = uc[idx], p2 = up[idx];
        vu = c2 + mom * (c2 - p2);                 // u_tmp
      }
      sx[rr][cc] = vx;
      su[rr][cc] = vu;
    }
    __syncthreads();
    const float* wa = &swa[f * KK];
    const float* wb = &swb[f * KK];
#pragma unroll
    for (int dp = 0; dp < 7; ++dp)
#pragma unroll
      for (int dq = 0; dq < 7; ++dq) {
        float w1 = wa[(6 - dp) * 7 + (6 - dq)];
        float w2 = wb[(6 - dp) * 7 + (6 - dq)];
        acc = fmaf(w1, sx[ty + dp][tx + dq], acc);
        acc = fmaf(w2, su[ty + dp][tx + dq], acc);
      }
  }
  long oi = (long)n * HY * WY + (long)(r0 + ty) * WY + (c0 + tx);
  res[oi] = y[oi] - acc;
}

// =====================================================================
// Kernel U: implicit-GEMM conv(res, [A;B]) via V_WMMA_F32_16X16X4_F32
// + fused FISTA update (momentum base, /L step, soft-threshold for u).
// WG = 256 threads (8 waves); covers 8 rows x 64 cols x 64 filters.
// M = 16 pixels of one row, N = 16 filters, K = 49 padded to 52 (13x K4).
// All 13 A-fragments are preloaded before the WMMA chain so the LDS
// round-trips overlap and the 13 WMMAs issue back-to-back.
// Writes x_new/u_new in place into the (dead) prev buffers.
// =====================================================================
__global__ __launch_bounds__(256) void densae_grad_update(
    const float* __restrict__ res,
    const float* __restrict__ A, const float* __restrict__ B,
    const float* __restrict__ xc, float* __restrict__ xpd,
    const float* __restrict__ uc, float* __restrict__ upd,
    float mom)
{
  __shared__ float sres[14][72];      // res tile (8+6) x (64+6), padded stride
  __shared__ float sw[52][64];        // filters [k][f], k>=49 zero padded
  __shared__ float sdt[8][16][17];    // per-wave D-tile transpose buffer

  const int tid = threadIdx.x;
  const int n  = blockIdx.z;
  const int r0 = blockIdx.y * 8;
  const int c0 = blockIdx.x * 64;
  const long ybase = (long)n * HY * WY;

  for (int i = tid; i < 14 * 70; i += 256) {
    int rr = i / 70, cc = i % 70;
    int gi = min(r0 + rr, HY - 1);           // clamp only feeds predicated-out pixels
    int gj = min(c0 + cc, WY - 1);
    sres[rr][cc] = res[ybase + (long)gi * WY + gj];
  }
  for (int i = tid; i < 52 * 64; i += 256) {
    int k = i >> 6, f = i & 63;
    float w = 0.f;
    if (k < KK) w = (f < 32) ? A[f * KK + k] : B[(f - 32) * KK + k];
    sw[k][f] = w;
  }
  __syncthreads();

  const int wid  = tid >> 5;
  const int lane = tid & 31;
  const int half = lane >> 4;     // lanes 16..31 hold K+2 / M+8 slots
  const int ln   = lane & 15;

  for (int nt = 0; nt < 4; ++nt) {
    // B fragments for this 16-filter group (held in VGPRs for all 13 K-steps)
    v2f bf[13];
#pragma unroll
    for (int s = 0; s < 13; ++s) {
      int k0 = 4 * s + 2 * half;
      bf[s].x = sw[k0][nt * 16 + ln];
      bf[s].y = sw[k0 + 1][nt * 16 + ln];
    }
    for (int j = 0; j < 4; ++j) {           // literal trip count: fully uniform
      const int mt = wid + 8 * j;
      const int rr = mt >> 2;      // row 0..7 within tile
      const int cs = mt & 3;       // 16-col segment 0..3
      const int cb = cs * 16 + ln;

      // Preload all A fragments for this tile, then chain the WMMAs.
      v2f af[13];
#pragma unroll
      for (int s = 0; s < 13; ++s) {
        int ka = 4 * s + 2 * half; if (ka > 48) ka = 48;  // k>48 has zero weight
        int kb = ka + 1;           if (kb > 48) kb = 48;
        af[s].x = sres[rr + ka / 7][cb + ka % 7];
        af[s].y = sres[rr + kb / 7][cb + kb % 7];
      }
      v8f acc = {0.f, 0.f, 0.f, 0.f, 0.f, 0.f, 0.f, 0.f};
#pragma unroll
      for (int s = 0; s < 13; ++s)
        acc = __builtin_amdgcn_wmma_f32_16x16x4_f32(
            false, af[s], false, bf[s], (short)0, acc, false, false);

      // Transpose D tile through per-wave LDS (same-wave LDS is in-order)
      // so epilogue loads/stores are coalesced along the column dimension.
#pragma unroll
      for (int v = 0; v < 8; ++v)
        sdt[wid][v + 8 * half][ln] = acc[v];    // [M][N]
      const int row  = r0 + rr;
      const int colb = c0 + cs * 16;
#pragma unroll
      for (int i = 0; i < 8; ++i) {
        int fl = i + 8 * half;                  // filter within N-tile
        int fg = nt * 16 + fl;                  // global filter 0..63
        int col = colb + ln;
        float d = sdt[wid][ln][fl];             // M = ln, N = fl
        if (row < HX && col < WX) {
          if (fg < 32) {                        // A dictionary -> x
            long idx = ((long)(n * CH + fg)) * HX * WX + (long)row * WX + col;
            float c1 = xc[idx], p1 = xpd[idx];
            float t1 = c1 + mom * (c1 - p1);    // x_tmp
            xpd[idx] = t1 + d * INV_L;          // x_new (in place into prev buf)
          } else {                              // B dictionary -> u
            long idx = ((long)(n * CH + (fg - 32))) * HX * WX + (long)row * WX + col;
            float c2 = uc[idx], p2 = upd[idx];
            float t2 = c2 + mom * (c2 - p2);    // u_tmp
            float v2 = t2 + d * INV_L;
            float mag = fabsf(v2) - THR;        // two-sided soft threshold
            v2 = copysignf(fmaxf(mag, 0.f), v2);
            upd[idx] = v2;
          }
        }
      }
    }
  }
}

// =====================================================================
// Kernel F: Ax_hat = convT(x,A); Bu_hat = convT(u,B); y_hat = Ax + Bu
// =====================================================================
__global__ __launch_bounds__(256) void densae_final(
    float* __restrict__ yhat, float* __restrict__ axh, float* __restrict__ buh,
    const float* __restrict__ x, const float* __restrict__ u,
    const float* __restrict__ A, const float* __restrict__ B)
{
  __shared__ float swa[CH * KK];
  __shared__ float swb[CH * KK];
  __shared__ float sx[22][24];
  __shared__ float su[22][24];

  const int tx = threadIdx.x, ty = threadIdx.y;
  const int tid = ty * 16 + tx;
  const int n  = blockIdx.z;
  const int r0 = blockIdx.y * 16;
  const int c0 = blockIdx.x * 16;

  for (int i = tid; i < CH * KK; i += 256) { swa[i] = A[i]; swb[i] = B[i]; }

  const long nbase = (long)n * CH * HX * WX;
  float accA = 0.f, accB = 0.f;

  for (int f = 0; f < CH; ++f) {
    __syncthreads();
    const long cb = nbase + (long)f * HX * WX;
    for (int i = tid; i < 22 * 22; i += 256) {
      int rr = i / 22, cc = i % 22;
      int gi = r0 - 6 + rr, gj = c0 - 6 + cc;
      float vx = 0.f, vu = 0.f;
      if (gi >= 0 && gi < HX && gj >= 0 && gj < WX) {
        long idx = cb + (long)gi * WX + gj;
        vx = x[idx];
        vu = u[idx];
      }
      sx[rr][cc] = vx;
      su[rr][cc] = vu;
    }
    __syncthreads();
    const float* wa = &swa[f * KK];
    const float* wb = &swb[f * KK];
#pragma unroll
    for (int dp = 0; dp < 7; ++dp)
#pragma unroll
      for (int dq = 0; dq < 7; ++dq) {
        float w1 = wa[(6 - dp) * 7 + (6 - dq)];
        float w2 = wb[(6 - dp) * 7 + (6 - dq)];
        accA = fmaf(w1, sx[ty + dp][tx + dq], accA);
        accB = fmaf(w2, su[ty + dp][tx + dq], accB);
      }
  }
  long oi = (long)n * HY * WY + (long)(r0 + ty) * WY + (c0 + tx);
  axh[oi]  = accA;
  buh[oi]  = accB;
  yhat[oi] = accA + accB;
}

// =====================================================================
// Host launcher
// =====================================================================
extern "C" void kernel_launch(void* const* d_in, const int* in_sizes, int n_in,
                              void* d_out, int out_size, void* d_ws, size_t ws_size,
                              hipStream_t stream) {
  (void)in_sizes; (void)n_in; (void)out_size; (void)ws_size;
  const float* y = (const float*)d_in[0];
  const float* A = (const float*)d_in[1];
  const float* B = (const float*)d_in[2];

  // d_out layout: y_hat | x_new | u_new | Ax_hat | Bu_hat
  float* out  = (float*)d_out;
  float* yhat = out;
  float* Xa   = out + YSZ;        // final x lands here (iter 14 writes Xa)
  float* Ua   = Xa + XSZ;
  float* axh  = Ua + XSZ;
  float* buh  = axh + YSZ;

  // workspace: second ping-pong buffers + residual
  float* Xb  = (float*)d_ws;
  float* Ub  = Xb + XSZ;
  float* res = Ub + XSZ;          // needs (2*XSZ + YSZ)*4 ~ 260 MB

  hipMemsetAsync(Xa, 0, XSZ * sizeof(float), stream);
  hipMemsetAsync(Ua, 0, XSZ * sizeof(float), stream);
  hipMemsetAsync(Xb, 0, XSZ * sizeof(float), stream);
  hipMemsetAsync(Ub, 0, XSZ * sizeof(float), stream);

  // FISTA momentum coefficients: mom[k] = (t_{k-1}-1)/t_k, t_0 = 1
  float mom[NITER];
  mom[0] = 0.f;
  {
    double tj = 1.0;
    for (int k = 1; k < NITER; ++k) {
      double tn = 0.5 * (1.0 + sqrt(1.0 + 4.0 * tj * tj));
      mom[k] = (float)((tj - 1.0) / tn);
      tj = tn;
    }
  }

  dim3 gR(WY / 16, HY / 16, NB), bR(16, 16);
  dim3 gU(4, 32, NB);

  for (int k = 0; k < NITER; ++k) {
    float *xcur, *xprev, *ucur, *uprev;
    if ((k & 1) == 0) { xprev = Xa; xcur = Xb; uprev = Ua; ucur = Ub; }
    else              { xprev = Xb; xcur = Xa; uprev = Ub; ucur = Ua; }
    densae_resid<<<gR, bR, 0, stream>>>(res, y, A, B,
                                        xcur, xprev, ucur, uprev, mom[k]);
    densae_grad_update<<<gU, 256, 0, stream>>>(res, A, B,
                                               xcur, xprev, ucur, uprev, mom[k]);
    // x_new written into xprev buffer => roles swap next iteration
  }

  densae_final<<<gR, bR, 0, stream>>>(yhat, axh, buh, Xa, Ua, A, B);
}